// InteractionBlock_18339510354336
// MI455X (gfx1250) — compile-verified
//
#include <hip/hip_runtime.h>
#include <hip/hip_bf16.h>
#include <math.h>

// ---------------------------------------------------------------------------
// Fused PaiNN-style interaction block for MI455X (gfx1250, wave32, WMMA).
// f32 WMMA 16x16x4 everywhere (problem is bandwidth-bound; f32 keeps accuracy).
// Scatter via global_atomic_add_f32 into L2-resident 51MB accumulators.
// ---------------------------------------------------------------------------

typedef __attribute__((ext_vector_type(2))) float v2f;
typedef __attribute__((ext_vector_type(8))) float v8f;

#define NND 50000
#define NE  800000

__device__ __forceinline__ v8f wmma4(v2f a, v2f b, v8f c) {
  // (neg_a, A, neg_b, B, c_mod, C, reuse_a, reuse_b) — 8-arg probed pattern
  return __builtin_amdgcn_wmma_f32_16x16x4_f32(false, a, false, b, (short)0, c,
                                               false, false);
}

__device__ __forceinline__ float swishf(float x) {
  return x / (1.0f + __expf(-x));
}

// -------------------------- zero the accumulators --------------------------
__global__ void zero_kernel(float4* p, long n4) {
  long i = (long)blockIdx.x * blockDim.x + threadIdx.x;
  const long stride = (long)gridDim.x * blockDim.x;
  const float4 z = make_float4(0.f, 0.f, 0.f, 0.f);
  for (; i < n4; i += stride) p[i] = z;
}

// ------------------- node up-transform: s@Wup0, v_x@Wup1 --------------------
// up_s: [N][64]; up_v: 3 planes of [N][64] (x-major planes)
__global__ void up_kernel(const float* __restrict__ nf,
                          const float* __restrict__ W0,
                          const float* __restrict__ W1,
                          float* __restrict__ up_s,
                          float* __restrict__ up_v) {
  const int lane = threadIdx.x & 31, half = lane >> 4, l16 = lane & 15;
  const int wave = (int)((blockIdx.x * blockDim.x + threadIdx.x) >> 5);
  const int n0 = wave * 16;
  if (n0 >= NND) return;                      // wave-uniform exit only
  const int kb = 2 * half;
  const float* rowA = nf + (size_t)(n0 + l16) * 256;
  for (int nt = 0; nt < 4; ++nt) {
    v8f as_ = {}, ax = {}, ay = {}, az = {};
    const int col = nt * 16 + l16;
    for (int k0 = 0; k0 < 64; k0 += 4) {
      v2f a_s, a_x, a_y, a_z, b0, b1;
#pragma unroll
      for (int r = 0; r < 2; ++r) {
        const int k = k0 + r + kb;
        a_s[r] = rowA[k];
        a_x[r] = rowA[64 + k * 3 + 0];
        a_y[r] = rowA[64 + k * 3 + 1];
        a_z[r] = rowA[64 + k * 3 + 2];
        b0[r] = W0[k * 64 + col];
        b1[r] = W1[k * 64 + col];
      }
      as_ = wmma4(a_s, b0, as_);
      ax  = wmma4(a_x, b1, ax);
      ay  = wmma4(a_y, b1, ay);
      az  = wmma4(a_z, b1, az);
    }
#pragma unroll
    for (int r = 0; r < 8; ++r) {
      const size_t m = (size_t)(n0 + r + 8 * half);
      up_s[m * 64 + col] = as_[r];
      up_v[(size_t)0 * NND * 64 + m * 64 + col] = ax[r];
      up_v[(size_t)1 * NND * 64 + m * 64 + col] = ay[r];
      up_v[(size_t)2 * NND * 64 + m * 64 + col] = az[r];
    }
  }
}

// ------------- fused edge kernel: radial MLP + messages + scatter ----------
__global__ void edge_kernel(const float* __restrict__ ef,     // E x 4
                            const float* __restrict__ rad,    // E x 8
                            const int* __restrict__ snd_g,
                            const int* __restrict__ rcv_g,
                            const unsigned char* __restrict__ msk_g,
                            const float* __restrict__ Wr1,    // 8 x 64
                            const float* __restrict__ Wr2,    // 64 x 64
                            const float* __restrict__ Wr3,    // 64 x 320
                            const float* __restrict__ up_s,
                            const float* __restrict__ up_v,
                            float* __restrict__ agg_s,
                            float* __restrict__ agg_v) {
  __shared__ float stg[8][16 * 68];           // per-wave 16x64 staging, padded
  const int lane = threadIdx.x & 31, half = lane >> 4, l16 = lane & 15;
  const int wv = threadIdx.x >> 5;
  const int e0 = (blockIdx.x * 8 + wv) * 16;  // 16 edges per wave
  float* st = stg[wv];
  const int kb = 2 * half;

  // per-reg edge metadata (C-layout row index m = r + 8*half)
  float4 efv[8];
  float mskv[8];
  int snd[8], rcv[8];
#pragma unroll
  for (int r = 0; r < 8; ++r) {
    const int e = e0 + r + 8 * half;
    efv[r] = ((const float4*)ef)[e];          // .x=sh0, .y/.z/.w=sh1
    mskv[r] = msk_g[e] ? 0.0625f : 0.0f;      // fold mask * (1/AVG_NEIGH)
    snd[r] = snd_g[e];
    rcv[r] = rcv_g[e];
  }

  // ---- Stage 1: h1 = swish(rad @ Wr1), K=8 ----
  const float* radRow = rad + (size_t)(e0 + l16) * 8;
  {
    v8f h[4];
    for (int nt = 0; nt < 4; ++nt) {
      v8f acc = {};
      for (int k0 = 0; k0 < 8; k0 += 4) {
        v2f a, b;
#pragma unroll
        for (int r = 0; r < 2; ++r) {
          const int k = k0 + r + kb;
          a[r] = radRow[k];
          b[r] = Wr1[k * 64 + nt * 16 + l16];
        }
        acc = wmma4(a, b, acc);
      }
#pragma unroll
      for (int r = 0; r < 8; ++r) acc[r] = swishf(acc[r]);
      h[nt] = acc;
    }
    __syncthreads();
#pragma unroll
    for (int nt = 0; nt < 4; ++nt)
#pragma unroll
      for (int r = 0; r < 8; ++r)
        st[(r + 8 * half) * 68 + nt * 16 + l16] = h[nt][r];  // D-layout -> [m][k]
    __syncthreads();
  }

  // ---- Stage 2: h2 = swish(h1 @ Wr2), K=64 ----
  {
    v8f h[4];
    for (int nt = 0; nt < 4; ++nt) {
      v8f acc = {};
      for (int k0 = 0; k0 < 64; k0 += 4) {
        v2f a, b;
#pragma unroll
        for (int r = 0; r < 2; ++r) {
          const int k = k0 + r + kb;
          a[r] = st[l16 * 68 + k];            // A-layout read
          b[r] = Wr2[k * 64 + nt * 16 + l16];
        }
        acc = wmma4(a, b, acc);
      }
#pragma unroll
      for (int r = 0; r < 8; ++r) acc[r] = swishf(acc[r]);
      h[nt] = acc;
    }
    __syncthreads();
#pragma unroll
    for (int nt = 0; nt < 4; ++nt)
#pragma unroll
      for (int r = 0; r < 8; ++r)
        st[(r + 8 * half) * 68 + nt * 16 + l16] = h[nt][r];
    __syncthreads();
  }

  // ---- Stage 3: per 16-channel chunk: 5 path tiles of R, fuse messages ----
  for (int j = 0; j < 4; ++j) {
    v8f Racc[5] = {{}, {}, {}, {}, {}};
    for (int k0 = 0; k0 < 64; k0 += 4) {
      v2f a;
#pragma unroll
      for (int r = 0; r < 2; ++r) a[r] = st[l16 * 68 + k0 + r + kb];
#pragma unroll
      for (int p = 0; p < 5; ++p) {           // share A across 5 WMMAs
        v2f b;
#pragma unroll
        for (int r = 0; r < 2; ++r)
          b[r] = Wr3[(k0 + r + kb) * 320 + p * 64 + j * 16 + l16];
        Racc[p] = wmma4(a, b, Racc[p]);
      }
    }
    const int colj = j * 16 + l16;
#pragma unroll
    for (int r = 0; r < 8; ++r) {
      const size_t s = (size_t)snd[r];
      const float sE = up_s[s * 64 + colj];
      const float vx = up_v[(size_t)0 * NND * 64 + s * 64 + colj];
      const float vy = up_v[(size_t)1 * NND * 64 + s * 64 + colj];
      const float vz = up_v[(size_t)2 * NND * 64 + s * 64 + colj];
      const float sh0 = efv[r].x, s1x = efv[r].y, s1y = efv[r].z, s1z = efv[r].w;
      const float mk = mskv[r];
      const float dot = vx * s1x + vy * s1y + vz * s1z;
      const float ms  = (Racc[0][r] * sE * sh0 + Racc[1][r] * dot) * mk;
      const float cx = vy * s1z - vz * s1y;
      const float cy = vz * s1x - vx * s1z;
      const float cz = vx * s1y - vy * s1x;
      const float mvx = (Racc[2][r] * sE * s1x + Racc[3][r] * vx * sh0 + Racc[4][r] * cx) * mk;
      const float mvy = (Racc[2][r] * sE * s1y + Racc[3][r] * vy * sh0 + Racc[4][r] * cy) * mk;
      const float mvz = (Racc[2][r] * sE * s1z + Racc[3][r] * vz * sh0 + Racc[4][r] * cz) * mk;
      const size_t d = (size_t)rcv[r];
      atomicAdd(agg_s + d * 64 + colj, ms);
      atomicAdd(agg_v + (size_t)0 * NND * 64 + d * 64 + colj, mvx);
      atomicAdd(agg_v + (size_t)1 * NND * 64 + d * 64 + colj, mvy);
      atomicAdd(agg_v + (size_t)2 * NND * 64 + d * 64 + colj, mvz);
    }
  }
}

// -------------- node down-transform: agg_s@Wdn0, agg_v@Wdn1 ----------------
__global__ void down_kernel(const float* __restrict__ agg_s,
                            const float* __restrict__ agg_v,
                            const float* __restrict__ Wd0,
                            const float* __restrict__ Wd1,
                            float* __restrict__ out) {
  const int lane = threadIdx.x & 31, half = lane >> 4, l16 = lane & 15;
  const int wave = (int)((blockIdx.x * blockDim.x + threadIdx.x) >> 5);
  const int n0 = wave * 16;
  if (n0 >= NND) return;
  const int kb = 2 * half;
  const size_t rowA = (size_t)(n0 + l16) * 64;
  for (int nt = 0; nt < 4; ++nt) {
    v8f as_ = {}, ax = {}, ay = {}, az = {};
    const int col = nt * 16 + l16;
    for (int k0 = 0; k0 < 64; k0 += 4) {
      v2f a_s, a_x, a_y, a_z, b0, b1;
#pragma unroll
      for (int r = 0; r < 2; ++r) {
        const int k = k0 + r + kb;
        a_s[r] = agg_s[rowA + k];
        a_x[r] = agg_v[(size_t)0 * NND * 64 + rowA + k];
        a_y[r] = agg_v[(size_t)1 * NND * 64 + rowA + k];
        a_z[r] = agg_v[(size_t)2 * NND * 64 + rowA + k];
        b0[r] = Wd0[k * 64 + col];
        b1[r] = Wd1[k * 64 + col];
      }
      as_ = wmma4(a_s, b0, as_);
      ax  = wmma4(a_x, b1, ax);
      ay  = wmma4(a_y, b1, ay);
      az  = wmma4(a_z, b1, az);
    }
#pragma unroll
    for (int r = 0; r < 8; ++r) {
      const size_t m = (size_t)(n0 + r + 8 * half);
      float* o = out + m * 256;
      o[col] = as_[r];
      o[64 + col * 3 + 0] = ax[r];
      o[64 + col * 3 + 1] = ay[r];
      o[64 + col * 3 + 2] = az[r];
    }
  }
}

// ---------------------------------------------------------------------------
extern "C" void kernel_launch(void* const* d_in, const int* in_sizes, int n_in,
                              void* d_out, int out_size, void* d_ws, size_t ws_size,
                              hipStream_t stream) {
  (void)in_sizes; (void)n_in; (void)out_size; (void)ws_size;
  const float* nf   = (const float*)d_in[0];
  const float* ef   = (const float*)d_in[1];
  const float* rad  = (const float*)d_in[2];
  const int*   snd  = (const int*)d_in[3];
  const int*   rcv  = (const int*)d_in[4];
  const unsigned char* msk = (const unsigned char*)d_in[5];
  const float* Wup0 = (const float*)d_in[6];
  const float* Wup1 = (const float*)d_in[7];
  const float* Wdn0 = (const float*)d_in[8];
  const float* Wdn1 = (const float*)d_in[9];
  const float* Wr1  = (const float*)d_in[10];
  const float* Wr2  = (const float*)d_in[11];
  const float* Wr3  = (const float*)d_in[12];
  float* out = (float*)d_out;

  // workspace: up_s | up_v(3 planes) | agg_s | agg_v(3 planes) = N*64*8 floats
  float* ws    = (float*)d_ws;
  float* up_s  = ws;
  float* up_v  = ws + (size_t)NND * 64;
  float* agg_s = ws + (size_t)NND * 64 * 4;
  float* agg_v = ws + (size_t)NND * 64 * 5;

  // zero aggregation buffers (N*64*4 floats -> float4 count)
  zero_kernel<<<2048, 256, 0, stream>>>((float4*)agg_s, (long)NND * 64);

  const int nodeWaves  = NND / 16;                    // 3125 (exact)
  const int nodeBlocks = (nodeWaves + 7) / 8;         // 8 waves / block
  up_kernel<<<nodeBlocks, 256, 0, stream>>>(nf, Wup0, Wup1, up_s, up_v);

  edge_kernel<<<NE / 128, 256, 0, stream>>>(ef, rad, snd, rcv, msk,
                                            Wr1, Wr2, Wr3, up_s, up_v,
                                            agg_s, agg_v);

  down_kernel<<<nodeBlocks, 256, 0, stream>>>(agg_s, agg_v, Wdn0, Wdn1, out);
}